// BaseNeuron_41652592836751
// MI455X (gfx1250) — compile-verified
//
#include <hip/hip_runtime.h>
#include <cstdint>

// LIF spiking-neuron scan, [T=64, B=16, N=65536] fp32 -> fp32 spikes.
//   h = v + (x - v)*0.5 ; s = (h >= 1) ; v = s ? 0 : h   (sequential over T)
//
// Roofline: 256 MB in + 256 MB out, ~0.5 flop/byte -> pure HBM stream,
// floor ~22us at 23.3 TB/s. T is a nonlinear recurrence, so parallelism is
// capped at B*N/4 = 262,144 threads = 8192 wave32. At ~800ns HBM latency the
// load half (11.65 TB/s) needs ~9.3 MB in flight; per-wave MLP must supply it.
// Depth-8 async global->LDS ring: 7 x 512B in flight per wave = ~28 MB total,
// comfortably above the latency-bandwidth product. v lives in VGPRs for all
// 64 steps; spikes leave via b128 non-temporal stores (no reuse; 512 MB
// stream >> 192 MB L2).

typedef __attribute__((ext_vector_type(4))) float v4f;

#define TAU_INV 0.5f
#define VTH 1.0f

// GLOBAL_LOAD_ASYNC_TO_LDS_B128: vdst = LDS byte address (low 32 bits of the
// flat shared pointer == LDS offset), vaddr = 64-bit global address, saddr=off.
__device__ __forceinline__ void async_ld_b128(unsigned lds_byte_addr, const void* gaddr) {
  asm volatile("global_load_async_to_lds_b128 %0, %1, off"
               :: "v"(lds_byte_addr), "v"(gaddr)
               : "memory");
}

template <int N>
__device__ __forceinline__ void wait_asynccnt() {
  asm volatile("s_wait_asynccnt %0" :: "i"(N) : "memory");
}

__device__ __forceinline__ v4f lif_step(v4f& v, v4f dv) {
  v4f s;
#pragma unroll
  for (int k = 0; k < 4; ++k) {
    float h = v[k] + (dv[k] - v[k]) * TAU_INV;  // neuronal_charge, tau=2 (exact *0.5)
    const bool fire = h >= VTH;                 // Heaviside fire
    s[k] = fire ? 1.0f : 0.0f;
    v[k] = fire ? 0.0f : h;                     // hard reset, v_reset = 0
  }
  return s;
}

// Template-recursive pipeline steps so every s_wait_asynccnt immediate is a
// true integer-constant-expression (required by the "i" asm constraint).
template <int T, int BLOCK, int D, long long STRIDE_F>
struct Pipe {
  template <int t>
  static __device__ __forceinline__ void step(v4f& v, v4f (*buf)[BLOCK], int tid,
                                              unsigned lds_base,
                                              const float* __restrict__ gx,
                                              float* __restrict__ gy) {
    if constexpr (t + D - 1 < T) {
      // Issue load t+D-1 into ring slot; D loads now unconsumed. In-order
      // completion => ASYNCcnt <= D-1 implies load t has landed in LDS.
      async_ld_b128(lds_base + ((t + D - 1) % D) * (BLOCK * 16),
                    gx + (long long)(t + D - 1) * STRIDE_F);
      wait_asynccnt<D - 1>();
    } else {
      // Tail: loads t..T-1 still unconsumed -> wait ASYNCcnt <= T-1-t.
      wait_asynccnt<T - 1 - t>();
    }
    v4f dv = buf[t % D][tid];   // ds_load_b128 (own slot only: no barrier)
    v4f s = lif_step(v, dv);
    __builtin_nontemporal_store(s, (v4f*)(gy + (long long)t * STRIDE_F));
    if constexpr (t + 1 < T) step<t + 1>(v, buf, tid, lds_base, gx, gy);
  }
};

template <int T, int BLOCK, int D, long long STRIDE_F>
__global__ __launch_bounds__(BLOCK)
void lif_scan_kernel(const float* __restrict__ x, float* __restrict__ out) {
  __shared__ v4f buf[D][BLOCK];  // D x 4KB ring per block (32 KB of 320 KB/WGP)

  const int tid = threadIdx.x;
  const long long i4 = (long long)blockIdx.x * BLOCK + tid;  // float4 index in [B*N]

  // Low 32 bits of the flat shared address are the LDS byte offset
  // (flat-LDS aperture).
  const unsigned lds_base = (unsigned)(uintptr_t)&buf[0][tid];

  const float* gx = x + 4 * i4;
  float* gy = out + 4 * i4;

  // Prologue: fill D-1 ring slots (loads t = 0..D-2).
#pragma unroll
  for (int j = 0; j < D - 1; ++j)
    async_ld_b128(lds_base + j * (BLOCK * 16), gx + (long long)j * STRIDE_F);

  v4f v = {0.0f, 0.0f, 0.0f, 0.0f};  // v0 = 0
  Pipe<T, BLOCK, D, STRIDE_F>::template step<0>(v, buf, tid, lds_base, gx, gy);
}

extern "C" void kernel_launch(void* const* d_in, const int* in_sizes, int n_in,
                              void* d_out, int out_size, void* d_ws, size_t ws_size,
                              hipStream_t stream) {
  (void)n_in; (void)d_ws; (void)ws_size; (void)out_size;
  const float* x = (const float*)d_in[0];
  float* out = (float*)d_out;

  constexpr int T = 64;                      // timesteps
  constexpr int BLOCK = 256;                 // 8 wave32 per block
  constexpr int D = 8;                       // async pipeline depth
  constexpr long long STRIDE_F = 16LL * 65536LL;  // B*N floats per timestep plane

  const int total = in_sizes[0];             // T*B*N = 67,108,864
  const int m4 = (total / T) / 4;            // float4 lanes = 262,144
  const int grid = m4 / BLOCK;               // 1024 blocks (exact)

  lif_scan_kernel<T, BLOCK, D, STRIDE_F><<<grid, BLOCK, 0, stream>>>(x, out);
}